// ScaledDotProductAttention_86122684219412
// MI455X (gfx1250) — compile-verified
//
#include <hip/hip_runtime.h>
#include <hip/hip_bf16.h>
#include <stdint.h>

// ---------------------------------------------------------------------------
// ScaledDotProductAttention (diag-score variant), MI455X / gfx1250.
// B=4096, N=256, D=128. Memory-bound (AI ~0.33 FLOP/B): optimize data movement.
// CDNA5 path used: Tensor Data Mover DMA of V[b] (256x128 f32, 128KB) into LDS
// overlapped with the Q.K streaming pass + softmax; s_wait_tensorcnt sync.
// ---------------------------------------------------------------------------

typedef float  f4    __attribute__((ext_vector_type(4)));
typedef unsigned int u32x4 __attribute__((ext_vector_type(4)));
typedef int    i32x4 __attribute__((ext_vector_type(4)));
typedef int    i32x8 __attribute__((ext_vector_type(8)));

#define BATCH 4096
#define NNEI  256
#define DDIM  128
#define INV_TEMP 0.08838834764831845f   // 1/sqrt(128)
#define MASK_FILL_V -1e10f

extern "C" __global__ __launch_bounds__(256)
void sdpa_diag_kernel(const float* __restrict__ q,
                      const float* __restrict__ k,
                      const float* __restrict__ v,
                      const float* __restrict__ r_pri,
                      const unsigned char* __restrict__ mask,  // jnp.bool_ = 1 byte
                      float* __restrict__ out,                 // [B, D]
                      float* __restrict__ attn_out)            // [B, N]
{
  // Dynamic LDS layout (base offset 0 => trivial D#.lds_addr):
  //   [0,                NNEI*DDIM)  : V tile (TDM destination), 32768 floats
  //   [NNEI*DDIM,        +NNEI)      : attn vector
  //   [NNEI*DDIM+NNEI,   +NNEI)      : reduction scratch
  extern __shared__ float smem[];
  float* sm_v   = smem;
  float* sm_att = smem + (NNEI * DDIM);
  float* sm_red = sm_att + NNEI;

  const int tid  = threadIdx.x;          // 0..255
  const int wave = tid >> 5;
  const int b    = blockIdx.x;

  // ---------------- Phase 0: kick off TDM DMA of V[b] into LDS -------------
  // One wave issues the tensor op; TENSORcnt is per-wave, so wave 0 also does
  // the s_wait_tensorcnt later, and a workgroup barrier releases the others.
  if (wave == 0) {
    uint64_t gaddr = (uint64_t)(uintptr_t)(v + (size_t)b * NNEI * DDIM);
    uint32_t g_lo = __builtin_amdgcn_readfirstlane((uint32_t)gaddr);
    uint32_t g_hi = __builtin_amdgcn_readfirstlane((uint32_t)(gaddr >> 32));

    // D# group 0 (128b): count=1 | lds_addr=0 | global_addr | type=2
    u32x4 g0;
    g0[0] = 1u;                                   // count=1, user descriptor
    g0[1] = 0u;                                   // lds_addr = 0 (dyn-LDS base)
    g0[2] = g_lo;                                 // global_addr[31:0]
    g0[3] = (g_hi & 0x01FFFFFFu) | 0x80000000u;   // addr[56:32] | type=2<<30

    // D# group 1 (256b): 2D tensor, data_size=4B, tile 128x256, stride 128
    i32x8 g1;
    g1[0] = (int)(2u << 16);                      // wg_mask=0 | data_size=2 (4B)
    g1[1] = (int)((uint32_t)DDIM << 16);          // tensor_dim0[15:0] in [31:16]
    g1[2] = (int)((uint32_t)NNEI << 16);          // dim0[31:16]=0 | tensor_dim1[15:0]
    g1[3] = (int)((uint32_t)DDIM << 16);          // dim1[31:16]=0 | tile_dim0=128
    g1[4] = (int)NNEI;                            // tile_dim1=256 | tile_dim2=0
    g1[5] = (int)DDIM;                            // tensor_dim0_stride = 128
    g1[6] = 0;                                    // stride0 hi | stride1 lo
    g1[7] = 0;                                    // stride1 hi

    i32x4 z4 = {0, 0, 0, 0};
#if __has_builtin(__builtin_amdgcn_tensor_load_to_lds)
#if __clang_major__ >= 23
    i32x8 z8 = {0, 0, 0, 0, 0, 0, 0, 0};
    __builtin_amdgcn_tensor_load_to_lds(g0, g1, z4, z4, z8, 0);
#else
    __builtin_amdgcn_tensor_load_to_lds(g0, g1, z4, z4, 0);
#endif
#endif
  }

  // ---------------- Phase 1: s = diag(Q.K^T) for row n = tid ----------------
  // Each thread owns one neighbor row: 32 x b128 loads from q and k, dot in
  // registers. No cross-lane reduction needed. Overlaps with the TDM of V.
  const f4* __restrict__ qr = (const f4*)(q + ((size_t)b * NNEI + tid) * DDIM);
  const f4* __restrict__ kr = (const f4*)(k + ((size_t)b * NNEI + tid) * DDIM);
  float acc = 0.0f;
#pragma unroll 8
  for (int j = 0; j < DDIM / 4; ++j) {
    f4 a = qr[j];
    f4 c = kr[j];
    acc += a.x * c.x + a.y * c.y + a.z * c.z + a.w * c.w;
  }

  // ---------------- Phase 2: scale, mask, softmax over N=256 ---------------
  float t = acc * r_pri[(size_t)b * NNEI + tid] * INV_TEMP;
  if (mask[(size_t)b * NNEI + tid]) t = MASK_FILL_V;

  // block max (tree reduce in LDS; sm_red is disjoint from the TDM region)
  sm_red[tid] = t;
  __syncthreads();
#pragma unroll
  for (int s = 128; s > 0; s >>= 1) {
    if (tid < s) sm_red[tid] = fmaxf(sm_red[tid], sm_red[tid + s]);
    __syncthreads();
  }
  const float row_max = sm_red[0];
  __syncthreads();

  const float e = __expf(t - row_max);
  sm_red[tid] = e;
  __syncthreads();
#pragma unroll
  for (int s = 128; s > 0; s >>= 1) {
    if (tid < s) sm_red[tid] = sm_red[tid] + sm_red[tid + s];
    __syncthreads();
  }
  const float inv_sum = 1.0f / sm_red[0];
  __syncthreads();

  const float a_n = e * inv_sum;
  attn_out[(size_t)b * NNEI + tid] = a_n;   // second output of the reference
  sm_att[tid] = a_n;

  // ---------------- Phase 3: out[b,:] = attn . V[b] from LDS ---------------
  // Wave 0 waits for its TDM; barrier releases everyone.
  if (wave == 0) {
    __builtin_amdgcn_s_wait_tensorcnt(0);
  }
  __syncthreads();

  // 256 threads: lane d = tid&127 accumulates half of N, halves combined.
  const int d    = tid & (DDIM - 1);
  const int half = tid >> 7;                 // 0 or 1
  float od = 0.0f;
#pragma unroll 8
  for (int i = 0; i < NNEI / 2; ++i) {
    const int n = half * (NNEI / 2) + i;
    od += sm_att[n] * sm_v[n * DDIM + d];    // attn: broadcast; V: stride-1 banks
  }
  if (half == 1) sm_red[d] = od;
  __syncthreads();
  if (half == 0) out[(size_t)b * DDIM + d] = od + sm_red[d];
}

extern "C" void kernel_launch(void* const* d_in, const int* in_sizes, int n_in,
                              void* d_out, int out_size, void* d_ws, size_t ws_size,
                              hipStream_t stream) {
  (void)in_sizes; (void)n_in; (void)out_size; (void)d_ws; (void)ws_size;
  const float*         q    = (const float*)d_in[0];
  const float*         k    = (const float*)d_in[1];
  const float*         v    = (const float*)d_in[2];
  const float*         r    = (const float*)d_in[3];
  const unsigned char* m    = (const unsigned char*)d_in[4];

  float* out  = (float*)d_out;                       // [B, D] first
  float* attn = (float*)d_out + (size_t)BATCH * DDIM; // then [B, N]

  const size_t smem_bytes = (size_t)(NNEI * DDIM + NNEI + NNEI) * sizeof(float); // 130 KB

  // Allow >64KB dynamic LDS (gfx1250 WGP has 320KB).
  (void)hipFuncSetAttribute((const void*)sdpa_diag_kernel,
                            hipFuncAttributeMaxDynamicSharedMemorySize,
                            (int)smem_bytes);

  hipLaunchKernelGGL(sdpa_diag_kernel, dim3(BATCH), dim3(256), smem_bytes, stream,
                     q, k, v, r, m, out, attn);
}